// WeightedProcrustes_37546604101741
// MI455X (gfx1250) — compile-verified
//
#include <hip/hip_runtime.h>
#include <math.h>

// CDNA5 wave32; FP32 WMMA operands: A 16x4 -> 2 VGPRs, B 4x16 -> 2 VGPRs, C/D -> 8 VGPRs
typedef __attribute__((ext_vector_type(2))) float v2f;
typedef __attribute__((ext_vector_type(8))) float v8f;

#define WAVES 8
#define TPB   256
#define FPAD  8   // padded feature row (floats) per staged pair

// ---------------------------------------------------------------------------
// Stage 1: gather pairs + accumulate weighted moments with V_WMMA_F32_16X16X4_F32
// Moment tile (per wave, 16x16 f32 accumulator, top-left 4x5 used):
//   rows (A, M-dim):  a = (Y0, Y1, Y2, 1)        (+ zero pad row via clamped idx)
//   cols (B, N-dim):  b = (w*X0, w*X1, w*X2, w, |w|)  (+ zero pad col)
//   D[r][c] += sum_pairs a_r * b_c   (K = pairs, 4 per WMMA)
// ---------------------------------------------------------------------------
__global__ __launch_bounds__(TPB) void wp_reduce_kernel(
    const float* __restrict__ xyzs1,
    const float* __restrict__ xyzs2,
    const int*   __restrict__ pairs,
    const float* __restrict__ wts,
    float* __restrict__ partials,
    int M, int N, int nblk)
{
  const int b    = blockIdx.y;
  const int blk  = blockIdx.x;
  const int tid  = threadIdx.x;
  const int wave = tid >> 5;
  const int lane = tid & 31;

  __shared__ float shA[WAVES][32][FPAD];  // a-features per staged pair (padded)
  __shared__ float shB[WAVES][32][FPAD];  // b-features per staged pair (padded)
  __shared__ float shPart[WAVES][20];     // per-wave 4x5 moment partials

  const int stride = nblk * TPB;
  const int start  = blk * TPB + tid;
  const int iters  = (N + stride - 1) / stride;   // uniform across all threads

  const size_t pbase = (size_t)b * N;
  const size_t xbase = (size_t)b * M * 3;

  v8f acc0 = {};                         // ping-pong accumulators to break the
  v8f acc1 = {};                         // 8-deep WMMA RAW chain

  const int m   = lane & 15;             // A row / B col selector for this lane
  const int klo = (lane < 16) ? 0 : 2;   // K pair selector (ISA 32-bit A/B layout)
  const int ma  = (m < 4) ? m : 4;       // clamped read idx -> slot 4 is zero pad
  const int nb  = (m < 5) ? m : 5;       // clamped read idx -> slot 5 is zero pad

  // Zero the pad slots once; the loop only rewrites slots [0..3] / [0..4].
  // Each lane owns its own row; same-wave DS ordering makes this race-free.
  shA[wave][lane][4] = 0.0f;
  shB[wave][lane][5] = 0.0f;

  for (int it = 0; it < iters; ++it) {
    int  i     = start + it * stride;
    bool valid = (i < N);
    int  ic    = valid ? i : (N - 1);    // clamp: keep loads in-bounds, zero features

    int2  pp = ((const int2*)pairs)[pbase + ic];        // coalesced 8B index load
    float w  = wts[pbase + ic];
    const float* Xp = xyzs1 + xbase + (size_t)pp.x * 3; // L2-resident gathers
    const float* Yp = xyzs2 + xbase + (size_t)pp.y * 3;
    float x0 = Xp[0], x1 = Xp[1], x2 = Xp[2];
    float y0 = Yp[0], y1 = Yp[1], y2 = Yp[2];

    float vf = valid ? 1.0f : 0.0f;
    w *= vf;
    // Vectorized staging: one b128 for a, one b128 + b32 for b (32B row stride)
    *(float4*)&shA[wave][lane][0] = make_float4(y0 * vf, y1 * vf, y2 * vf, vf);
    *(float4*)&shB[wave][lane][0] = make_float4(w * x0, w * x1, w * x2, w);
    shB[wave][lane][4] = fabsf(w);
    // Producer/consumer within one wave on a private LDS slice: DS ops are
    // in-order per wave; compiler inserts s_wait_dscnt. No barrier needed.

    #pragma unroll
    for (int g = 0; g < 8; ++g) {        // 8 WMMAs x K=4 pairs = 32 staged pairs
      int k0 = g * 4 + klo;
      v2f av, bv;
      av.x = shA[wave][k0    ][ma];      // unconditional ds loads (pad slots = 0)
      av.y = shA[wave][k0 + 1][ma];
      bv.x = shB[wave][k0    ][nb];
      bv.y = shB[wave][k0 + 1][nb];
      // D = A(16x4) x B(4x16) + C, full-EXEC, f32 accumulate
      if (g & 1)
        acc1 = __builtin_amdgcn_wmma_f32_16x16x4_f32(
            false, av, false, bv, (short)0, acc1, false, false);
      else
        acc0 = __builtin_amdgcn_wmma_f32_16x16x4_f32(
            false, av, false, bv, (short)0, acc0, false, false);
    }
  }

  v8f acc = acc0 + acc1;

  // Extract top-left 4x5 of the tile: lane n<16 holds D[r][n] in acc[r] (r=0..7)
  if (lane < 5) {
    #pragma unroll
    for (int r = 0; r < 4; ++r) shPart[wave][r * 5 + lane] = acc[r];
  }
  __syncthreads();

  // Deterministic fixed-order cross-wave sum -> one 20-float partial per block
  if (tid < 20) {
    float s = 0.0f;
    for (int wv = 0; wv < WAVES; ++wv) s += shPart[wv][tid];
    partials[((size_t)b * nblk + blk) * 20 + tid] = s;
  }
}

// ---------------------------------------------------------------------------
// Stage 2: per-batch finalize. Sum block partials (fixed order), reconstruct
// Sxy, then Horn's quaternion method (== SVD + reflection fix), write R, t, ws.
// ---------------------------------------------------------------------------
__global__ void wp_finalize_kernel(const float* __restrict__ partials,
                                   float* __restrict__ out,
                                   int nblk, int B)
{
  const int b   = blockIdx.x;
  const int tid = threadIdx.x;
  __shared__ float mom[20];

  if (tid < 20) {
    float s = 0.0f;
    for (int k = 0; k < nblk; ++k)
      s += partials[((size_t)b * nblk + k) * 20 + tid];
    mom[tid] = s;
  }
  __syncthreads();
  if (tid != 0) return;

  const float EPS = 1.1920929e-07f;
  // mom[r*5+c]: r in {Y0,Y1,Y2,1}, c in {wX0,wX1,wX2,w,|w|}
  float W1p = mom[19] + EPS;            // sum|w| + eps
  float inv = 1.0f / W1p;
  float mux[3] = { mom[15] * inv, mom[16] * inv, mom[17] * inv };
  float muy[3] = { mom[3]  * inv, mom[8]  * inv, mom[13] * inv };
  float sfac = mom[18] * inv - 2.0f;    // (sum w') - 2

  // Sxy = sum w'(Y-muy)(X-mux)^T = M1 + (s-2) muy mux^T
  float S[3][3];
  for (int i = 0; i < 3; ++i)
    for (int j = 0; j < 3; ++j)
      S[i][j] = mom[i * 5 + j] * inv + sfac * muy[i] * mux[j];

  // Horn correlation Mh[a][b] = sum x_a y_b = S[b][a]
  float Sxx = S[0][0], Sxy = S[1][0], Sxz = S[2][0];
  float Syx = S[0][1], Syy = S[1][1], Syz = S[2][1];
  float Szx = S[0][2], Szy = S[1][2], Szz = S[2][2];

  float N4[4][4];
  N4[0][0] =  Sxx + Syy + Szz; N4[0][1] = Syz - Szy; N4[0][2] = Szx - Sxz; N4[0][3] = Sxy - Syx;
  N4[1][1] =  Sxx - Syy - Szz; N4[1][2] = Sxy + Syx; N4[1][3] = Szx + Sxz;
  N4[2][2] = -Sxx + Syy - Szz; N4[2][3] = Syz + Szy;
  N4[3][3] = -Sxx - Syy + Szz;
  N4[1][0] = N4[0][1]; N4[2][0] = N4[0][2]; N4[3][0] = N4[0][3];
  N4[2][1] = N4[1][2]; N4[3][1] = N4[1][3]; N4[3][2] = N4[2][3];

  // Gershgorin shift so the max eigenvalue dominates under power iteration
  float sig = 0.0f;
  for (int i = 0; i < 4; ++i) {
    float rs = 0.0f;
    for (int j = 0; j < 4; ++j) rs += fabsf(N4[i][j]);
    sig = fmaxf(sig, rs);
  }
  for (int i = 0; i < 4; ++i) N4[i][i] += sig;

  float q[4] = {1.0f, 0.01f, 0.02f, 0.03f};
  for (int iter = 0; iter < 64; ++iter) {
    float r[4];
    for (int i = 0; i < 4; ++i)
      r[i] = N4[i][0]*q[0] + N4[i][1]*q[1] + N4[i][2]*q[2] + N4[i][3]*q[3];
    float nrm = sqrtf(r[0]*r[0] + r[1]*r[1] + r[2]*r[2] + r[3]*r[3]) + 1e-30f;
    float in = 1.0f / nrm;
    q[0] = r[0]*in; q[1] = r[1]*in; q[2] = r[2]*in; q[3] = r[3]*in;
  }

  float q0 = q[0], q1 = q[1], q2 = q[2], q3 = q[3];
  float R[3][3];
  R[0][0] = 1.0f - 2.0f*(q2*q2 + q3*q3);
  R[0][1] = 2.0f*(q1*q2 - q0*q3);
  R[0][2] = 2.0f*(q1*q3 + q0*q2);
  R[1][0] = 2.0f*(q1*q2 + q0*q3);
  R[1][1] = 1.0f - 2.0f*(q1*q1 + q3*q3);
  R[1][2] = 2.0f*(q2*q3 - q0*q1);
  R[2][0] = 2.0f*(q1*q3 - q0*q2);
  R[2][1] = 2.0f*(q2*q3 + q0*q1);
  R[2][2] = 1.0f - 2.0f*(q1*q1 + q2*q2);

  for (int i = 0; i < 3; ++i)
    for (int j = 0; j < 3; ++j)
      out[b * 9 + i * 3 + j] = R[i][j];

  for (int i = 0; i < 3; ++i) {
    float ti = muy[i] - (R[i][0]*mux[0] + R[i][1]*mux[1] + R[i][2]*mux[2]);
    out[B * 9 + b * 3 + i] = ti;
  }
  out[B * 12 + b] = mom[18];            // ws = sum w
}

// ---------------------------------------------------------------------------
extern "C" void kernel_launch(void* const* d_in, const int* in_sizes, int n_in,
                              void* d_out, int out_size, void* d_ws, size_t ws_size,
                              hipStream_t stream)
{
  const float* xyzs1 = (const float*)d_in[0];
  const float* xyzs2 = (const float*)d_in[1];
  const int*   pairs = (const int*)d_in[2];
  const float* wts   = (const float*)d_in[3];
  float* out      = (float*)d_out;
  float* partials = (float*)d_ws;

  const int B = 16;
  const int M = in_sizes[0] / (B * 3);
  const int N = in_sizes[2] / (B * 2);

  int nblk = 64;  // blocks per batch; scratch = B*nblk*20 floats (80 KB)
  size_t need = (size_t)B * nblk * 20 * sizeof(float);
  if (ws_size > 0 && need > ws_size) {
    nblk = (int)(ws_size / ((size_t)B * 20 * sizeof(float)));
    if (nblk < 1) nblk = 1;
  }

  dim3 grid(nblk, B);
  wp_reduce_kernel<<<grid, TPB, 0, stream>>>(xyzs1, xyzs2, pairs, wts,
                                             partials, M, N, nblk);
  wp_finalize_kernel<<<B, 32, 0, stream>>>(partials, out, nblk, B);
}